// BreakoutPlatformPosition_86517821213192
// MI455X (gfx1250) — compile-verified
//
#include <hip/hip_runtime.h>
#include <hip/hip_bf16.h>
#include <stdint.h>

// Problem geometry (from the reference): obs shape (16, 32, 3, 208, 160) f32.
#define WCOLS       160
#define ROW_STRIDE  (3 * 208 * 160)   // floats per (bs, oc) sample
#define ROW_OFFSET  (188 * 160)       // channel 0, platform row 188

// Address-space-typed pointers for the gfx1250 async-DMA builtin
// (builtin signature: AS(1) int* src, AS(3) int* dst, imm offset, imm cpol).
typedef __attribute__((address_space(1))) int glb_int;   // global
typedef __attribute__((address_space(3))) int lds_int;   // LDS

// One dword per lane, global -> LDS, tracked on ASYNCcnt (CDNA5 path).
__device__ __forceinline__ void async_copy_b32(const float* gsrc, float* ldst) {
  // Low 32 bits of a generic shared pointer are the LDS byte offset
  uint32_t loff = (uint32_t)(uintptr_t)ldst;
#if __has_builtin(__builtin_amdgcn_global_load_async_to_lds_b32)
  __builtin_amdgcn_global_load_async_to_lds_b32(
      (glb_int*)(uintptr_t)gsrc, (lds_int*)loff, /*offset=*/0, /*cpol=*/0);
#else
  asm volatile("global_load_async_to_lds_b32 %0, %1, off"
               :: "v"(loff), "v"((uint64_t)(uintptr_t)gsrc)
               : "memory");
#endif
}

__device__ __forceinline__ void wait_async0() {
#if __has_builtin(__builtin_amdgcn_s_wait_asynccnt)
  __builtin_amdgcn_s_wait_asynccnt(0);
  asm volatile("" ::: "memory");   // keep LDS reads below the wait
#else
  asm volatile("s_wait_asynccnt 0x0" ::: "memory");
#endif
}

__device__ __forceinline__ uint32_t ballot32(bool p) {
#if __has_builtin(__builtin_amdgcn_ballot_w32)
  return __builtin_amdgcn_ballot_w32(p);
#else
  return (uint32_t)__ballot(p);    // wave32: low 32 bits valid
#endif
}

// 192-bit value in three 64-bit words; r = a & (a >> s), s in (0, 64).
struct U192 { uint64_t x0, x1, x2; };
__device__ __forceinline__ U192 and_shr(const U192 a, const int s) {
  U192 r;
  r.x0 = a.x0 & ((a.x0 >> s) | (a.x1 << (64 - s)));
  r.x1 = a.x1 & ((a.x1 >> s) | (a.x2 << (64 - s)));
  r.x2 = a.x2 & (a.x2 >> s);
  return r;
}

// One wave (32 lanes) per (bs, oc) row. 512 single-wave workgroups total.
__global__ void __launch_bounds__(32)
BreakoutPlatformPosition_kernel(const float* __restrict__ obs,
                                float* __restrict__ out, int nrows) {
  __shared__ float row[WCOLS];
  const int rid  = blockIdx.x;
  const int lane = threadIdx.x;        // 0..31
  if (rid >= nrows) return;

  const float* g = obs + (size_t)rid * ROW_STRIDE + ROW_OFFSET;

  // Stage the 160-float row into LDS via the async DMA path.
  // 5 async b32 ops/wave: lane l moves columns l, l+32, ..., l+128.
  #pragma unroll
  for (int k = 0; k < 5; ++k) {
    const int c = lane + 32 * k;
    async_copy_b32(g + c, &row[c]);
  }
  wait_async0();

  // Threshold constants, rounded exactly like jnp.float32(py-float expr).
  const float lo = (float)(100.0 / 255.0 - 0.15);
  const float up = (float)(200.0 / 255.0 + 0.15);

  // Build the 160-bit mask with wave32 ballots: bit c == mask at column c.
  uint32_t w[5];
  #pragma unroll
  for (int k = 0; k < 5; ++k) {
    const float v = row[lane + 32 * k];
    w[k] = ballot32((v >= lo) & (v <= up));
  }
  w[4] &= 0x7FFFFFFFu;   // m[..., -1] = False (column 159)

  // 160-bit mask M, zero-extended to 192 bits.
  U192 M;
  M.x0 = (uint64_t)w[0] | ((uint64_t)w[1] << 32);
  M.x1 = (uint64_t)w[2] | ((uint64_t)w[3] << 32);
  M.x2 = (uint64_t)w[4];

  // Length-doubling run detector:
  //   R2[i]  = bits i..i+1 set,  R4[i] = i..i+3,  R8[i] = i..i+7,
  //   R12[i] = R8[i] & R4[i+8]  = bits i..i+11 all set.
  const U192 R2 = and_shr(M,  1);
  const U192 R4 = and_shr(R2, 2);
  const U192 R8 = and_shr(R4, 4);
  U192 R12;
  R12.x0 = R8.x0 & ((R4.x0 >> 8) | (R4.x1 << 56));
  R12.x1 = R8.x1 & ((R4.x1 >> 8) | (R4.x2 << 56));
  R12.x2 = R8.x2 & (R4.x2 >> 8);

  // First set bit of R12 == start of first run of >=12 == first_hit - 11.
  int pos;
  if (R12.x0)      pos = __builtin_ctzll(R12.x0);
  else if (R12.x1) pos = 64 + __builtin_ctzll(R12.x1);
  else if (R12.x2) pos = 128 + __builtin_ctzll(R12.x2);
  else             pos = -1;

  if (lane == 0) out[rid] = (float)pos;
}

extern "C" void kernel_launch(void* const* d_in, const int* in_sizes, int n_in,
                              void* d_out, int out_size, void* d_ws, size_t ws_size,
                              hipStream_t stream) {
  (void)n_in; (void)d_ws; (void)ws_size; (void)out_size;
  const float* obs = (const float*)d_in[0];
  float* out = (float*)d_out;
  const int nrows = in_sizes[0] / ROW_STRIDE;   // 16*32 = 512
  hipLaunchKernelGGL(BreakoutPlatformPosition_kernel,
                     dim3(nrows), dim3(32), 0, stream, obs, out, nrows);
}